// BITWISE_VAE_4389456577196
// MI455X (gfx1250) — compile-verified
//
#include <hip/hip_runtime.h>
#include <hip/hip_bf16.h>
#include <stdint.h>

#define Bb   128
#define Tt   4096
#define Cc   32
#define NSUM 5457                       // 1+16+64+256+1024+4096
#define QS   0.17677669529663687f       // 1/sqrt(32)
#define KK   70.71067811865476f         // 4*INV_TEMP/sqrt(32)

typedef float v8f  __attribute__((ext_vector_type(8)));
typedef float v2f  __attribute__((ext_vector_type(2)));
typedef unsigned int v4u __attribute__((ext_vector_type(4)));
typedef int   v8i  __attribute__((ext_vector_type(8)));
typedef int   v4i  __attribute__((ext_vector_type(4)));

#if defined(__has_builtin)
#if __has_builtin(__builtin_amdgcn_tensor_load_to_lds)
#define HAVE_TDM 1
#endif
#if __has_builtin(__builtin_amdgcn_wmma_f32_16x16x4_f32)
#define HAVE_WMMA_F32K4 1
#endif
#endif

struct UpArgs {
  const unsigned int* qb0; const unsigned int* qb1; const unsigned int* qb2;
  const unsigned int* qb3; const unsigned int* qb4;
  int n;   // number of previous scales to accumulate
};

// ---- piecewise-linear upsample of a sign-mask codebook, evaluated at fine t (lane = channel)
__device__ __forceinline__ float lerp_prev(const unsigned int* qb, int pt, float ratio,
                                           int b, int t, int lane) {
  float pos = fmaf((float)t + 0.5f, ratio, -0.5f);
  pos = fminf(fmaxf(pos, 0.0f), (float)(pt - 1));
  int lo = (int)pos;
  int hi = min(lo + 1, pt - 1);
  float w = pos - (float)lo;
  unsigned int ml = qb[b * pt + lo];
  unsigned int mh = qb[b * pt + hi];
  float ql = ((ml >> lane) & 1u) ? QS : -QS;
  float qh = ((mh >> lane) & 1u) ? QS : -QS;
  return fmaf(w, qh - ql, ql);
}

__device__ __forceinline__ float up_sum(const UpArgs& ua, int b, int t, int lane) {
  float s = 0.f;
  if (ua.n > 0) s += lerp_prev(ua.qb0, 1,    1.0f   / 4096.f, b, t, lane);
  if (ua.n > 1) s += lerp_prev(ua.qb1, 16,   16.0f  / 4096.f, b, t, lane);
  if (ua.n > 2) s += lerp_prev(ua.qb2, 64,   64.0f  / 4096.f, b, t, lane);
  if (ua.n > 3) s += lerp_prev(ua.qb3, 256,  256.0f / 4096.f, b, t, lane);
  if (ua.n > 4) s += lerp_prev(ua.qb4, 1024, 1024.0f/ 4096.f, b, t, lane);
  return s;
}

// ---- per-row sum of squares over C=32 for 16 rows, via V_WMMA_F32_16X16X4_F32 (B = ones)
// tile: 16 rows x 36-float padded stride in LDS (conflict-free for the A-layout gathers)
__device__ __forceinline__ void rowN2(const float r[16], float* tile, int lane, float n2[16]) {
#if defined(HAVE_WMMA_F32K4)
  #pragma unroll
  for (int m = 0; m < 16; ++m) tile[m * 36 + lane] = r[m];
  asm volatile("s_wait_dscnt 0" ::: "memory");
  v2f ones; ones.x = 1.f; ones.y = 1.f;
  v8f acc = {0.f,0.f,0.f,0.f,0.f,0.f,0.f,0.f};
  int mr = lane & 15, off2 = (lane >> 4) * 2;   // ISA A-layout: lanes 0-15 K0/K1, lanes 16-31 K2/K3
  #pragma unroll
  for (int kk = 0; kk < 8; ++kk) {
    v2f a;
    a.x = tile[mr * 36 + kk * 4 + off2];
    a.y = tile[mr * 36 + kk * 4 + off2 + 1];
    a.x *= a.x; a.y *= a.y;
    acc = __builtin_amdgcn_wmma_f32_16x16x4_f32(false, a, false, ones, (short)0, acc, false, false);
  }
  // D layout: VGPR i -> M=i (lanes 0-15) / M=8+i (lanes 16-31); any N column holds the row sum
  #pragma unroll
  for (int m = 0; m < 16; ++m) n2[m] = __shfl(acc[m & 7], (m < 8) ? 0 : 16, 32);
#else
  #pragma unroll
  for (int m = 0; m < 16; ++m) {
    float v = r[m] * r[m];
    #pragma unroll
    for (int k = 16; k >= 1; k >>= 1) v += __shfl_xor(v, k, 32);
    n2[m] = v;
  }
#endif
}

// ---- workgroup-level reduction of loss statistics, then fp64 atomics
// acc layout (doubles): [0..5] entSum, [6..11] commitSum, [12..12+191] probSum[scale][c]
__device__ __forceinline__ void commitStats(float (*red)[34], int wv, int lane,
                                            float entA, float comA, float prA,
                                            double* acc, int j) {
  #pragma unroll
  for (int m = 16; m >= 1; m >>= 1) {
    entA += __shfl_xor(entA, m, 32);
    comA += __shfl_xor(comA, m, 32);
  }
  red[wv][lane] = prA;
  if (lane == 0) { red[wv][32] = entA; red[wv][33] = comA; }
  __syncthreads();
  if (wv == 0) {
    float pv = 0.f;
    #pragma unroll
    for (int k = 0; k < 8; ++k) pv += red[k][lane];
    atomicAdd(&acc[12 + j * 32 + lane], (double)pv);
    if (lane < 2) {
      float tv = 0.f;
      #pragma unroll
      for (int k = 0; k < 8; ++k) tv += red[k][32 + lane];
      atomicAdd(&acc[(lane == 0) ? j : (6 + j)], (double)tv);
    }
  }
}

// ================= kernels =================

__global__ void k_init(double* acc, float* D0) {
  int i = threadIdx.x;
  for (int k = i; k < 204; k += 256) acc[k] = 0.0;
  for (int k = i; k < Bb * Cc; k += 256) D0[k] = 0.f;
}

// Streaming downsample of the on-the-fly residual to coarse row SUMS.
// 1024 waves: wave = (b, 512-t chunk). f tiles (32t x 32c = 4 KiB) DMA'd into LDS by the TDM,
// double buffered, synced with s_wait_tensorcnt.
__global__ void __launch_bounds__(256) k_down(const float* __restrict__ f,
                                              float* __restrict__ D,
                                              UpArgs ua, int pt, int bsLog) {
#if defined(HAVE_TDM)
  __shared__ float ftile[8][2][1024];       // 64 KiB
#endif
  int wv = threadIdx.x >> 5, lane = threadIdx.x & 31;
  int w = blockIdx.x * 8 + wv;              // 0..1023
  int b = w >> 3, chunk = w & 7;
  int t0 = chunk << 9;
  int BS = 1 << bsLog;

#if defined(HAVE_TDM)
  auto tdm = [&](int tile, int buf) {
    unsigned int ldsOff = (unsigned int)(uintptr_t)(&ftile[wv][buf][0]);
    unsigned long long ga =
        (unsigned long long)(uintptr_t)(f + ((size_t)(b * Tt + t0 + tile * 32)) * Cc);
    v4u g0;
    g0.x = 1u;  // count=1, user descriptor
    g0.y = (unsigned int)__builtin_amdgcn_readfirstlane((int)ldsOff);
    g0.z = (unsigned int)__builtin_amdgcn_readfirstlane((int)(unsigned int)(ga & 0xffffffffull));
    g0.w = (unsigned int)__builtin_amdgcn_readfirstlane(
        (int)(((unsigned int)(ga >> 32) & 0x01ffffffu) | 0x80000000u));  // type=2 ("image")
    // group1: data_size=4B; tensor_dim0=tile_dim0=stride=1024 elems; tile_dim1=tensor_dim1=1
    v8i g1 = {0x00020000, 0x04000000, 0x00010000, 0x04000000, 1, 1024, 0x04000000, 0};
    v4i z4 = {0, 0, 0, 0};
    v8i z8 = {0, 0, 0, 0, 0, 0, 0, 0};
    __builtin_amdgcn_tensor_load_to_lds(g0, g1, z4, z4, z8, 0);
  };
  tdm(0, 0);
#endif
  float acc = 0.f;
  for (int tile = 0; tile < 16; ++tile) {
#if defined(HAVE_TDM)
    if (tile < 15) { tdm(tile + 1, (tile + 1) & 1); __builtin_amdgcn_s_wait_tensorcnt(1); }
    else           { __builtin_amdgcn_s_wait_tensorcnt(0); }
    asm volatile("" ::: "memory");
    const float* tb = &ftile[wv][tile & 1][0];
#endif
    for (int tl = 0; tl < 32; ++tl) {
      int t = t0 + tile * 32 + tl;
#if defined(HAVE_TDM)
      float fv = tb[tl * 32 + lane];
#else
      float fv = f[((size_t)(b * Tt + t)) * Cc + lane];
#endif
      acc += fv - up_sum(ua, b, t, lane);
      if (bsLog <= 9 && ((t & (BS - 1)) == BS - 1)) {
        D[((size_t)(b * pt + (t >> bsLog))) * Cc + lane] = acc;
        acc = 0.f;
      }
    }
  }
  if (bsLog > 9) atomicAdd(&D[b * Cc + lane], acc);   // pt==1: partial sums across 8 chunks
}

// BSQ on coarse row-sums: normalize (WMMA norm), sign-quantize, pack mask, stats.
__global__ void __launch_bounds__(256) k_bsq_coarse(const float* __restrict__ D,
                                                    unsigned int* __restrict__ qbits,
                                                    float* __restrict__ outBits,  // pre-offset
                                                    double* __restrict__ acc,
                                                    int j, int pt, float invBS, int numTiles) {
  __shared__ float tileSh[8][16 * 36];
  __shared__ float red[8][34];
  int wv = threadIdx.x >> 5, lane = threadIdx.x & 31;
  int wgl = blockIdx.x * 8 + wv;
  int nW = gridDim.x * 8;
  float entA = 0.f, comA = 0.f, prA = 0.f;
  for (int tl = wgl; tl < numTiles; tl += nW) {
    int row0 = tl * 16;
    float r[16], n2[16];
    #pragma unroll
    for (int m = 0; m < 16; ++m)
      r[m] = D[((size_t)(row0 + m)) * Cc + lane] * invBS;
    rowN2(r, &tileSh[wv][0], lane, n2);
    #pragma unroll
    for (int m = 0; m < 16; ++m) {
      float den = fmaxf(sqrtf(n2[m]), 1e-12f);
      float fn  = r[m] / den;
      bool  bit = fn > 0.f;
      float zh  = bit ? QS : -QS;
      int row = row0 + m;
      int bb = row / pt, tc = row - bb * pt;
      __builtin_nontemporal_store(bit ? 1.f : 0.f,
                                  &outBits[((size_t)bb * NSUM + tc) * Cc + lane]);
      unsigned int msk = (unsigned int)__ballot(bit);
      if (lane == 0) qbits[row] = msk;
      float p = 1.f / (1.f + __expf(KK * fn));
      entA += -(p * __logf(p + 1e-8f) + (1.f - p) * __logf(1.f - p + 1e-8f));
      prA  += p;
      comA += (zh - fn) * (zh - fn);
    }
  }
  commitStats(red, wv, lane, entA, comA, prA, acc, j);
}

// Final full-resolution scale: residual on the fly, BSQ, quantized = L + zhat, bits, stats.
__global__ void __launch_bounds__(256) k_bsq_full(const float* __restrict__ f, UpArgs ua,
                                                  float* __restrict__ outQ,
                                                  float* __restrict__ outBits, // pre-offset
                                                  double* __restrict__ acc) {
  __shared__ float tileSh[8][16 * 36];
  __shared__ float red[8][34];
  int wv = threadIdx.x >> 5, lane = threadIdx.x & 31;
  int wgl = blockIdx.x * 8 + wv;
  int nW = gridDim.x * 8;
  const int numTiles = Bb * Tt / 16;   // 32768
  float entA = 0.f, comA = 0.f, prA = 0.f;
  for (int tl = wgl; tl < numTiles; tl += nW) {
    int row0 = tl * 16;
    int b = row0 >> 12;
    int t0 = row0 & (Tt - 1);
    float r[16], L[16], n2[16];
    #pragma unroll
    for (int m = 0; m < 16; ++m) {
      int t = t0 + m;
      L[m] = up_sum(ua, b, t, lane);
      float fv = f[((size_t)(b * Tt + t)) * Cc + lane];
      r[m] = fv - L[m];
    }
    rowN2(r, &tileSh[wv][0], lane, n2);
    #pragma unroll
    for (int m = 0; m < 16; ++m) {
      float den = fmaxf(sqrtf(n2[m]), 1e-12f);
      float fn  = r[m] / den;
      bool  bit = fn > 0.f;
      float zh  = bit ? QS : -QS;
      int t = t0 + m;
      __builtin_nontemporal_store(bit ? 1.f : 0.f,
                                  &outBits[((size_t)b * NSUM + t) * Cc + lane]);
      __builtin_nontemporal_store(L[m] + zh,
                                  &outQ[((size_t)(b * Tt + t)) * Cc + lane]);
      float p = 1.f / (1.f + __expf(KK * fn));
      entA += -(p * __logf(p + 1e-8f) + (1.f - p) * __logf(1.f - p + 1e-8f));
      prA  += p;
      comA += (zh - fn) * (zh - fn);
    }
  }
  commitStats(red, wv, lane, entA, comA, prA, acc, 5);
}

__global__ void k_loss(const double* __restrict__ acc, float* __restrict__ outL) {
  int lane = threadIdx.x & 31;
  const int pts[6] = {1, 16, 64, 256, 1024, 4096};
  for (int j = 0; j < 6; ++j) {
    float inv = 1.f / (float)(Bb * pts[j]);
    float q = (float)acc[12 + j * 32 + lane] * inv;
    float cbc = -(q * __logf(q + 1e-8f) + (1.f - q) * __logf(1.f - q + 1e-8f));
    #pragma unroll
    for (int m = 16; m >= 1; m >>= 1) cbc += __shfl_xor(cbc, m, 32);
    if (lane == 0) {
      float ps  = (float)acc[j] * inv;
      float cm  = (float)acc[6 + j] * inv;
      float pen = (ps - cbc) / 100.f;
      outL[j] = 0.1f * pen + 0.2f * cm;
    }
  }
}

// ================= host =================

extern "C" void kernel_launch(void* const* d_in, const int* in_sizes, int n_in,
                              void* d_out, int out_size, void* d_ws, size_t ws_size,
                              hipStream_t stream) {
  const float* f = (const float*)d_in[0];
  float* out = (float*)d_out;
  char* ws = (char*)d_ws;

  // workspace: D (coarse row sums, reused, 16 MiB) | packed sign masks | fp64 stat accumulators
  float* D = (float*)ws;
  unsigned int* qb0 = (unsigned int*)(ws + 16777216);
  unsigned int* qb1 = (unsigned int*)(ws + 16777216 + 512);
  unsigned int* qb2 = (unsigned int*)(ws + 16777216 + 512 + 8192);
  unsigned int* qb3 = (unsigned int*)(ws + 16777216 + 512 + 8192 + 32768);
  unsigned int* qb4 = (unsigned int*)(ws + 16777216 + 512 + 8192 + 32768 + 131072);
  double* acc = (double*)(ws + 17474048);

  float* outQ    = out;
  float* bitsBase = out + (size_t)Bb * Tt * Cc;               // 16777216
  float* outL    = out + (size_t)Bb * Tt * Cc + (size_t)Bb * NSUM * Cc;

  const int pts[5]   = {1, 16, 64, 256, 1024};
  const int bsLog[5] = {12, 8, 6, 4, 2};
  const int offs[6]  = {0, 1, 17, 81, 337, 1361};
  const int tiles[5] = {8, 128, 512, 2048, 8192};
  unsigned int* qbArr[5] = {qb0, qb1, qb2, qb3, qb4};

  k_init<<<1, 256, 0, stream>>>(acc, D);

  UpArgs ua{qb0, qb1, qb2, qb3, qb4, 0};
  for (int j = 0; j < 5; ++j) {
    ua.n = j;
    k_down<<<128, 256, 0, stream>>>(f, D, ua, pts[j], bsLog[j]);
    float invBS = 1.f / (float)(1 << bsLog[j]);
    int wgs = (tiles[j] + 7) / 8;
    k_bsq_coarse<<<wgs, 256, 0, stream>>>(D, qbArr[j],
                                          bitsBase + (size_t)offs[j] * Cc,
                                          acc, j, pts[j], invBS, tiles[j]);
  }
  ua.n = 5;
  k_bsq_full<<<2048, 256, 0, stream>>>(f, ua, outQ,
                                       bitsBase + (size_t)offs[5] * Cc, acc);
  k_loss<<<1, 32, 0, stream>>>(acc, outL);
}